// DiffAttn_86431921865310
// MI455X (gfx1250) — compile-verified
//
#include <hip/hip_runtime.h>
#include <hip/hip_bf16.h>
#include <math.h>

// ---------------------------------------------------------------------------
// Problem constants (reference: B=4, L=512, U=512, H=8, D=64)
// ---------------------------------------------------------------------------
#define U_DIM 512
#define B_DIM 4
#define L_DIM 512
#define H_DIM 8
#define D_DIM 64
#define HB_DIM 32        // H*B
#define M_ROWS 2048      // B*L

typedef __attribute__((ext_vector_type(2))) float v2f;
typedef __attribute__((ext_vector_type(8))) float v8f;

// V_WMMA_F32_16X16X4_F32: A=16x4 f32 (2 VGPR), B=4x16 f32 (2 VGPR), C/D=16x16 f32 (8 VGPR)
__device__ __forceinline__ v8f wmma4(v2f a, v2f b, v8f c) {
#if defined(__HIP_DEVICE_COMPILE__)
  return __builtin_amdgcn_wmma_f32_16x16x4_f32(false, a, false, b, (short)0, c,
                                               false, false);
#else
  (void)a; (void)b;
  return c;
#endif
}

// B-operand tiles are stored k-pair interleaved: element (k, col) lives at
// [k>>1][col*2 + (k&1)], so a B fragment (B[k][col], B[k+1][col]) is one
// 8-byte aligned LDS word -> single ds_load_b64 into an even VGPR pair.
// Pair-row strides are chosen == 32 (mod 64 banks) so the k+2 half-wave hits
// the disjoint half of the LDS banks.

// ---------------------------------------------------------------------------
// Kernel 1: fused projection GEMM + bias + ReLU, stored head-split.
//   Tile: 64(M) x 128(N) per 256-thread block; 8 waves (4 row x 2 col groups),
//   each wave 16 rows x 64 cols = 4 v8f accumulators.
// ---------------------------------------------------------------------------
#define SW_STRIDE 288  // 128 cols * 2 + pad; 288 % 64 == 32

__global__ __launch_bounds__(256) void proj_relu_kernel(
    const float* __restrict__ Xq, const float* __restrict__ Xk,
    const float* __restrict__ Xv, const float* __restrict__ Wq,
    const float* __restrict__ Wk, const float* __restrict__ Wv,
    const float* __restrict__ bq, const float* __restrict__ bk,
    const float* __restrict__ bv, float* __restrict__ Qh,
    float* __restrict__ Kh, float* __restrict__ Vh) {
  const int which = blockIdx.z;
  const float* X = which == 0 ? Xq : (which == 1 ? Xk : Xv);
  const float* W = which == 0 ? Wq : (which == 1 ? Wk : Wv);
  const float* bias = which == 0 ? bq : (which == 1 ? bk : bv);
  float* O = which == 0 ? Qh : (which == 1 ? Kh : Vh);

  const int m0 = blockIdx.y * 64;
  const int n0 = blockIdx.x * 128;
  const int tid = threadIdx.x;
  const int lane = tid & 31;
  const int wave = tid >> 5;
  const int wm = wave >> 1;  // row group (16 rows)
  const int wn = wave & 1;   // col group (64 cols)
  const int l16 = lane & 15;
  const int hi = lane >> 4;

  __shared__ float sX[64][68];            // [m][k], A-side (stride 68: 4i banks)
  __shared__ float sW2[8][SW_STRIDE];     // [k>>1][col*2 + (k&1)], B-side

  v8f acc[4] = {};

  for (int kt = 0; kt < U_DIM / 16; ++kt) {
    // Stage X tile 64x16 (each thread: one float4)
    {
      const int r = tid >> 2, c = (tid & 3) * 4;
      const float4 x =
          *(const float4*)&X[(size_t)(m0 + r) * U_DIM + kt * 16 + c];
      *(float4*)&sX[r][c] = x;
    }
    // Stage W tile 16x128 interleaved (each thread: 8 cols of one k row)
    {
      const int r = tid >> 4, c = (tid & 15) * 8;
      const float4 w0 =
          *(const float4*)&W[(size_t)(kt * 16 + r) * U_DIM + n0 + c];
      const float4 w1 =
          *(const float4*)&W[(size_t)(kt * 16 + r) * U_DIM + n0 + c + 4];
      float* dst = &sW2[r >> 1][c * 2 + (r & 1)];
      dst[0] = w0.x;  dst[2] = w0.y;  dst[4] = w0.z;  dst[6] = w0.w;
      dst[8] = w1.x;  dst[10] = w1.y; dst[12] = w1.z; dst[14] = w1.w;
    }
    __syncthreads();

#pragma unroll
    for (int kk = 0; kk < 16; kk += 4) {
      const v2f a = *(const v2f*)&sX[wm * 16 + l16][kk + 2 * hi];
      const int kp = (kk >> 1) + hi;
#pragma unroll
      for (int n = 0; n < 4; ++n) {
        const int col = wn * 64 + n * 16 + l16;
        const v2f b = *(const v2f*)&sW2[kp][col * 2];
        acc[n] = wmma4(a, b, acc[n]);
      }
    }
    __syncthreads();
  }

  // Epilogue: bias + relu, write head-split [(h*B+b)*L + l]*D + d
#pragma unroll
  for (int n = 0; n < 4; ++n) {
    const int ng = n0 + wn * 64 + n * 16 + l16;  // global col = h*64 + d
    const int h = ng >> 6, d = ng & 63;
    const float bb = bias[ng];
#pragma unroll
    for (int r = 0; r < 8; ++r) {
      const int mg = m0 + wm * 16 + r + 8 * hi;  // global row = b*512 + l
      const int b = mg >> 9, l = mg & 511;
      float v = acc[n][r] + bb;
      v = v > 0.0f ? v : 0.0f;
      O[(((size_t)(h * B_DIM + b) * L_DIM + l) * D_DIM) + d] = v;
    }
  }
}

// ---------------------------------------------------------------------------
// Kernel 2: auxiliary reductions. qq=sum(Q^2,d), kk=sum(K^2,d),
//           qsign = sign(|sum_u queries|)
// ---------------------------------------------------------------------------
__global__ __launch_bounds__(256) void aux_kernel(
    const float* __restrict__ Qh, const float* __restrict__ Kh,
    const float* __restrict__ queries, float* __restrict__ qq,
    float* __restrict__ kkv, float* __restrict__ qsign) {
  const int idx = blockIdx.x * blockDim.x + threadIdx.x;
  const int nrows = HB_DIM * L_DIM;  // 16384
  if (idx < nrows) {
    const float* p = Qh + (size_t)idx * D_DIM;
    float s = 0.0f;
    for (int d = 0; d < D_DIM; ++d) s += p[d] * p[d];
    qq[idx] = s;
  } else if (idx < 2 * nrows) {
    const float* p = Kh + (size_t)(idx - nrows) * D_DIM;
    float s = 0.0f;
    for (int d = 0; d < D_DIM; ++d) s += p[d] * p[d];
    kkv[idx - nrows] = s;
  } else if (idx < 2 * nrows + B_DIM * L_DIM) {
    const int r = idx - 2 * nrows;
    const float* p = queries + (size_t)r * U_DIM;
    float s = 0.0f;
    for (int u = 0; u < U_DIM; ++u) s += p[u];
    qsign[r] = (s != 0.0f) ? 1.0f : 0.0f;
  }
}

// ---------------------------------------------------------------------------
// Kernel 3: fused euclidean attention.
//   block = (64 query rows) x (one head-batch). 8 waves.
//   Pass A: scores via WMMA, write masked attn_score (NT), online softmax.
//   Pass B: recompute scores, P=exp(s-m)/L, WMMA P*V, scale rows by qm.
// ---------------------------------------------------------------------------
#define SKT_STRIDE 96   // 32 kcols * 2 + pad; 96 % 64 == 32
#define SV_STRIDE 160   // 64 dcols * 2 + pad; 160 % 64 == 32

__global__ __launch_bounds__(256) void attn_kernel(
    const float* __restrict__ Qh, const float* __restrict__ Kh,
    const float* __restrict__ Vh, const float* __restrict__ qq,
    const float* __restrict__ kkv, const float* __restrict__ qsign,
    float* __restrict__ outp, float* __restrict__ ascore) {
  const int hb = blockIdx.y;       // 0..31
  const int q0 = blockIdx.x * 64;  // query tile base
  const int tid = threadIdx.x;
  const int lane = tid & 31;
  const int wave = tid >> 5;
  const int wm = wave >> 1;  // 0..3: 16-row group
  const int wn = wave & 1;   // 0..1: 16-key-col (pass A) / 32-d-col (pass B)
  const int l16 = lane & 15;
  const int hi = lane >> 4;

  __shared__ float sQ[64][68];             // [q][d]       A-side
  __shared__ float sKT2[32][SKT_STRIDE];   // [d>>1][k*2 + (d&1)]  B-side
  __shared__ float sV2[16][SV_STRIDE];     // [k>>1][d*2 + (k&1)]  B-side
  __shared__ float sP[64][36];             // [q][k]       A-side
  __shared__ float sKK[L_DIM];
  __shared__ float sQQ[64];
  __shared__ float sQM[64];
  __shared__ float sMrow[64];
  __shared__ float sInvL[64];
  __shared__ float sStat[2][64][2];

  // ---- stage Q tile, kk vector, qq, qm ----
  {
    const int r = tid >> 2, c = (tid & 3) * 16;
    const size_t base = ((size_t)hb * L_DIM + q0 + r) * D_DIM + c;
#pragma unroll
    for (int j = 0; j < 4; ++j) {
      const float4 v = *(const float4*)&Qh[base + 4 * j];
      *(float4*)&sQ[r][c + 4 * j] = v;
    }
    sKK[2 * tid] = kkv[(size_t)hb * L_DIM + 2 * tid];
    sKK[2 * tid + 1] = kkv[(size_t)hb * L_DIM + 2 * tid + 1];
    if (tid < 64) {
      sQQ[tid] = qq[(size_t)hb * L_DIM + q0 + tid];
      sQM[tid] = qsign[(size_t)(hb & 3) * L_DIM + q0 + tid];
    }
  }
  __syncthreads();

  // ---- Pass A: scores + online softmax stats ----
  float m_r[8], l_r[8];
#pragma unroll
  for (int r = 0; r < 8; ++r) { m_r[r] = -INFINITY; l_r[r] = 0.0f; }

  for (int kt = 0; kt < L_DIM / 32; ++kt) {
    {  // stage K tile transposed + interleaved: (kc, d) -> sKT2[d>>1][kc*2+(d&1)]
      const int kc = tid >> 3, d0 = (tid & 7) * 8;  // d0 even
      const size_t base = ((size_t)hb * L_DIM + kt * 32 + kc) * D_DIM + d0;
      const float4 k0 = *(const float4*)&Kh[base];
      const float4 k1 = *(const float4*)&Kh[base + 4];
      float* dst = &sKT2[d0 >> 1][kc * 2];  // 8B aligned, float2 stores
      *(v2f*)&dst[0] = v2f{k0.x, k0.y};
      *(v2f*)&dst[SKT_STRIDE] = v2f{k0.z, k0.w};
      *(v2f*)&dst[2 * SKT_STRIDE] = v2f{k1.x, k1.y};
      *(v2f*)&dst[3 * SKT_STRIDE] = v2f{k1.z, k1.w};
    }
    __syncthreads();

    v8f s = {};
#pragma unroll
    for (int kk = 0; kk < D_DIM; kk += 4) {
      const v2f a = *(const v2f*)&sQ[wm * 16 + l16][kk + 2 * hi];
      const v2f b = *(const v2f*)&sKT2[(kk >> 1) + hi][(wn * 16 + l16) * 2];
      s = wmma4(a, b, s);
    }

    const int kcol = wn * 16 + l16;
    const int kg = kt * 32 + kcol;
#pragma unroll
    for (int r = 0; r < 8; ++r) {
      const int qrow = wm * 16 + r + 8 * hi;
      float raw = sQQ[qrow] + sKK[kg] - 2.0f * s[r];
      raw = fmaxf(raw, 1e-12f);
      const float sc = -sqrtf(raw) * 0.125f;  // / sqrt(D=64)
      __builtin_nontemporal_store(
          sc * sQM[qrow],
          &ascore[((size_t)hb * L_DIM + q0 + qrow) * L_DIM + kg]);
      const float mn = fmaxf(m_r[r], sc);
      l_r[r] = l_r[r] * __expf(m_r[r] - mn) + __expf(sc - mn);
      m_r[r] = mn;
    }
    __syncthreads();
  }

  // merge stats across the 16 lanes holding the same row
#pragma unroll
  for (int r = 0; r < 8; ++r) {
    float m = m_r[r], l = l_r[r];
#pragma unroll
    for (int off = 8; off >= 1; off >>= 1) {
      const float mo = __shfl_xor(m, off, 16);
      const float lo = __shfl_xor(l, off, 16);
      const float mn = fmaxf(m, mo);
      l = l * __expf(m - mn) + lo * __expf(mo - mn);
      m = mn;
    }
    m_r[r] = m; l_r[r] = l;
  }
  if (l16 == 0) {
#pragma unroll
    for (int r = 0; r < 8; ++r) {
      const int row = wm * 16 + r + 8 * hi;
      sStat[wn][row][0] = m_r[r];
      sStat[wn][row][1] = l_r[r];
    }
  }
  __syncthreads();
  if (tid < 64) {
    const float ma = sStat[0][tid][0], la = sStat[0][tid][1];
    const float mb = sStat[1][tid][0], lb = sStat[1][tid][1];
    const float mn = fmaxf(ma, mb);
    const float L = la * __expf(ma - mn) + lb * __expf(mb - mn);
    sMrow[tid] = mn;
    sInvL[tid] = 1.0f / L;
  }
  __syncthreads();

  // ---- Pass B: recompute scores, P = exp(s-m)/L, accumulate P @ V ----
  v8f o[2] = {};
  for (int kt = 0; kt < L_DIM / 32; ++kt) {
    {  // stage K (transposed+interleaved) and V (interleaved) tiles
      const int kc = tid >> 3, d0 = (tid & 7) * 8;
      const size_t kb = ((size_t)hb * L_DIM + kt * 32 + kc) * D_DIM + d0;
      const float4 k0 = *(const float4*)&Kh[kb];
      const float4 k1 = *(const float4*)&Kh[kb + 4];
      float* dk = &sKT2[d0 >> 1][kc * 2];
      *(v2f*)&dk[0] = v2f{k0.x, k0.y};
      *(v2f*)&dk[SKT_STRIDE] = v2f{k0.z, k0.w};
      *(v2f*)&dk[2 * SKT_STRIDE] = v2f{k1.x, k1.y};
      *(v2f*)&dk[3 * SKT_STRIDE] = v2f{k1.z, k1.w};
      const float4 v0 = *(const float4*)&Vh[kb];
      const float4 v1 = *(const float4*)&Vh[kb + 4];
      float* dv = &sV2[kc >> 1][d0 * 2 + (kc & 1)];
      dv[0] = v0.x;  dv[2] = v0.y;  dv[4] = v0.z;  dv[6] = v0.w;
      dv[8] = v1.x;  dv[10] = v1.y; dv[12] = v1.z; dv[14] = v1.w;
    }
    __syncthreads();

    v8f s = {};
#pragma unroll
    for (int kk = 0; kk < D_DIM; kk += 4) {
      const v2f a = *(const v2f*)&sQ[wm * 16 + l16][kk + 2 * hi];
      const v2f b = *(const v2f*)&sKT2[(kk >> 1) + hi][(wn * 16 + l16) * 2];
      s = wmma4(a, b, s);
    }
    const int kcol = wn * 16 + l16;
    const int kg = kt * 32 + kcol;
#pragma unroll
    for (int r = 0; r < 8; ++r) {
      const int qrow = wm * 16 + r + 8 * hi;
      float raw = sQQ[qrow] + sKK[kg] - 2.0f * s[r];
      raw = fmaxf(raw, 1e-12f);
      const float sc = -sqrtf(raw) * 0.125f;
      sP[qrow][kcol] = __expf(sc - sMrow[qrow]) * sInvL[qrow];
    }
    __syncthreads();

    // O += P(64x32) @ V(32x64); this wave: rows wm*16.., d-cols wn*32..
#pragma unroll
    for (int kk = 0; kk < 32; kk += 4) {
      const v2f a = *(const v2f*)&sP[wm * 16 + l16][kk + 2 * hi];
      const int kp = (kk >> 1) + hi;
#pragma unroll
      for (int n = 0; n < 2; ++n) {
        const int dcol = wn * 32 + n * 16 + l16;
        const v2f b = *(const v2f*)&sV2[kp][dcol * 2];
        o[n] = wmma4(a, b, o[n]);
      }
    }
    __syncthreads();
  }

  // ---- epilogue: out[b][q][h*64+d] = qm * O ----
  const int h = hb >> 2, b = hb & 3;
#pragma unroll
  for (int n = 0; n < 2; ++n) {
    const int d = wn * 32 + n * 16 + l16;
#pragma unroll
    for (int r = 0; r < 8; ++r) {
      const int qrow = wm * 16 + r + 8 * hi;
      const float v = o[n][r] * sQM[qrow];
      __builtin_nontemporal_store(
          v, &outp[((size_t)b * L_DIM + q0 + qrow) * U_DIM + h * D_DIM + d]);
    }
  }
}

// ---------------------------------------------------------------------------
// Host launcher
// ---------------------------------------------------------------------------
extern "C" void kernel_launch(void* const* d_in, const int* in_sizes, int n_in,
                              void* d_out, int out_size, void* d_ws,
                              size_t ws_size, hipStream_t stream) {
  const float* queries = (const float*)d_in[0];
  const float* keys = (const float*)d_in[1];
  const float* values = (const float*)d_in[2];
  const float* Wq = (const float*)d_in[3];
  const float* bq = (const float*)d_in[4];
  const float* Wk = (const float*)d_in[5];
  const float* bk = (const float*)d_in[6];
  const float* Wv = (const float*)d_in[7];
  const float* bv = (const float*)d_in[8];

  float* outp = (float*)d_out;                           // [4,512,512]
  float* ascore = outp + (size_t)B_DIM * L_DIM * U_DIM;  // [32,512,512]

  float* ws = (float*)d_ws;
  float* Qh = ws;  // [32][512][64]
  float* Kh = Qh + (size_t)HB_DIM * L_DIM * D_DIM;
  float* Vh = Kh + (size_t)HB_DIM * L_DIM * D_DIM;
  float* qq = Vh + (size_t)HB_DIM * L_DIM * D_DIM;  // [32*512]
  float* kkv = qq + (size_t)HB_DIM * L_DIM;
  float* qsign = kkv + (size_t)HB_DIM * L_DIM;      // [4*512]

  // 1) projections (Q,K,V) with bias+relu, head-split layout
  proj_relu_kernel<<<dim3(U_DIM / 128, M_ROWS / 64, 3), 256, 0, stream>>>(
      queries, keys, values, Wq, Wk, Wv, bq, bk, bv, Qh, Kh, Vh);

  // 2) row norms + query mask
  const int aux_threads = 2 * HB_DIM * L_DIM + B_DIM * L_DIM;
  aux_kernel<<<(aux_threads + 255) / 256, 256, 0, stream>>>(Qh, Kh, queries,
                                                            qq, kkv, qsign);

  // 3) fused euclidean attention + softmax + P@V
  attn_kernel<<<dim3(L_DIM / 64, HB_DIM), 256, 0, stream>>>(
      Qh, Kh, Vh, qq, kkv, qsign, outp, ascore);
}